// ProjectedLinear_26645977104594
// MI455X (gfx1250) — compile-verified
//
#include <hip/hip_runtime.h>

typedef __attribute__((ext_vector_type(2))) float v2f;
typedef __attribute__((ext_vector_type(8))) float v8f;

#define D_LARGE 4096
#define D_SMALL 768
#define RANK    64
#define NROWS   8192                 // 4 * 2048
#define KSPLIT  8
#define KCHUNK  (D_LARGE / KSPLIT)   // 512

// ---------------- tiny precompute kernels (negligible cost) ----------------

// V = B @ W_small : V[r, j] = sum_k B[r,k] * W[k,j]    (64 x 768)
__global__ void k_bw(const float* __restrict__ B, const float* __restrict__ W,
                     float* __restrict__ V) {
  int idx = blockIdx.x * blockDim.x + threadIdx.x;     // 64*768 threads
  int r = idx / D_SMALL, j = idx % D_SMALL;
  float s = 0.f;
  for (int k = 0; k < D_SMALL; ++k)
    s = fmaf(B[r * D_SMALL + k], W[k * D_SMALL + j], s);
  V[idx] = s;
}

// M[a,b] = dot(B[a,:], V[b,:])   == (B @ W_small.T @ B.T)[a,b]   (64 x 64)
__global__ void k_m(const float* __restrict__ B, const float* __restrict__ V,
                    float* __restrict__ M) {
  int idx = blockIdx.x * blockDim.x + threadIdx.x;     // 64*64 threads
  int a = idx / RANK, b = idx % RANK;
  float s = 0.f;
  for (int j = 0; j < D_SMALL; ++j)
    s = fmaf(B[a * D_SMALL + j], V[b * D_SMALL + j], s);
  M[idx] = s;
}

// C[d, rin] = sum_rout A[d,rout] * M[rin,rout]  (4096 x 64); also At = A.T
__global__ void k_c(const float* __restrict__ A, const float* __restrict__ M,
                    float* __restrict__ C, float* __restrict__ At) {
  int idx = blockIdx.x * blockDim.x + threadIdx.x;     // 4096*64 threads
  int d = idx / RANK, rin = idx % RANK;
  float s = 0.f;
  for (int r = 0; r < RANK; ++r)
    s = fmaf(A[d * RANK + r], M[rin * RANK + r], s);
  C[d * RANK + rin] = s;
  At[rin * D_LARGE + d] = A[d * RANK + rin];
}

// ---------------- main GEMM 1: T1 = x @ A  (8192 x 4096 @ 4096 x 64) -------
// Per wave: 32-row x 64-col tile (two 16-row strips share B-fragments);
// K split across gridDim.y; partials written to P, reduced deterministically.
__global__ __launch_bounds__(256) void k_xa(const float* __restrict__ x,
                                            const float* __restrict__ At,
                                            float* __restrict__ P) {
  const int lane    = threadIdx.x & 31;
  const int wv      = threadIdx.x >> 5;
  const int rowbase = blockIdx.x * 256 + wv * 32;
  const int k0      = blockIdx.y * KCHUNK;
  const int l15     = lane & 15;
  const int khalf   = (lane >> 4) << 1;   // lanes 16-31 carry K+2,K+3

  const float* __restrict__ xr0 = x  + (size_t)(rowbase + l15) * D_LARGE;
  const float* __restrict__ xr1 = xr0 + (size_t)16 * D_LARGE;
  const float* __restrict__ b0p = At + (size_t)(l15 +  0) * D_LARGE;
  const float* __restrict__ b1p = At + (size_t)(l15 + 16) * D_LARGE;
  const float* __restrict__ b2p = At + (size_t)(l15 + 32) * D_LARGE;
  const float* __restrict__ b3p = At + (size_t)(l15 + 48) * D_LARGE;

  v8f a00 = {}, a01 = {}, a02 = {}, a03 = {};
  v8f a10 = {}, a11 = {}, a12 = {}, a13 = {};

#pragma unroll 2
  for (int k = k0; k < k0 + KCHUNK; k += 4) {
    const int kk = k + khalf;
    v2f af0 = *(const v2f*)(xr0 + kk);
    v2f af1 = *(const v2f*)(xr1 + kk);
    v2f bf0 = *(const v2f*)(b0p + kk);
    v2f bf1 = *(const v2f*)(b1p + kk);
    v2f bf2 = *(const v2f*)(b2p + kk);
    v2f bf3 = *(const v2f*)(b3p + kk);
    a00 = __builtin_amdgcn_wmma_f32_16x16x4_f32(false, af0, false, bf0, (short)0, a00, false, false);
    a01 = __builtin_amdgcn_wmma_f32_16x16x4_f32(false, af0, false, bf1, (short)0, a01, false, false);
    a02 = __builtin_amdgcn_wmma_f32_16x16x4_f32(false, af0, false, bf2, (short)0, a02, false, false);
    a03 = __builtin_amdgcn_wmma_f32_16x16x4_f32(false, af0, false, bf3, (short)0, a03, false, false);
    a10 = __builtin_amdgcn_wmma_f32_16x16x4_f32(false, af1, false, bf0, (short)0, a10, false, false);
    a11 = __builtin_amdgcn_wmma_f32_16x16x4_f32(false, af1, false, bf1, (short)0, a11, false, false);
    a12 = __builtin_amdgcn_wmma_f32_16x16x4_f32(false, af1, false, bf2, (short)0, a12, false, false);
    a13 = __builtin_amdgcn_wmma_f32_16x16x4_f32(false, af1, false, bf3, (short)0, a13, false, false);
  }

  // D layout: VGPR v -> row v + 8*(lane>=16), col = lane&15 within 16x16 tile
  float* __restrict__ p0 = P + ((size_t)blockIdx.y * NROWS + rowbase + (lane >> 4) * 8) * RANK + l15;
  float* __restrict__ p1 = p0 + (size_t)16 * RANK;
#pragma unroll
  for (int v = 0; v < 8; ++v) {
    p0[(size_t)v * RANK +  0] = a00[v];
    p0[(size_t)v * RANK + 16] = a01[v];
    p0[(size_t)v * RANK + 32] = a02[v];
    p0[(size_t)v * RANK + 48] = a03[v];
    p1[(size_t)v * RANK +  0] = a10[v];
    p1[(size_t)v * RANK + 16] = a11[v];
    p1[(size_t)v * RANK + 32] = a12[v];
    p1[(size_t)v * RANK + 48] = a13[v];
  }
}

// Deterministic K-split reduction: T1 = sum_ks P[ks]
__global__ void k_red(const float* __restrict__ P, float* __restrict__ T1) {
  int i = blockIdx.x * blockDim.x + threadIdx.x;       // NROWS*RANK threads
  float s = 0.f;
#pragma unroll
  for (int ks = 0; ks < KSPLIT; ++ks)
    s += P[(size_t)ks * NROWS * RANK + i];
  T1[i] = s;
}

// ---------------- main GEMM 2: out = T1 @ C.T (8192 x 64 @ 64 x 4096) ------
// Per wave: 32-row x 64-col tile, K = 64 fully unrolled.
__global__ __launch_bounds__(256) void k_out(const float* __restrict__ T1,
                                             const float* __restrict__ C,
                                             float* __restrict__ out) {
  const int lane    = threadIdx.x & 31;
  const int wv      = threadIdx.x >> 5;
  const int rowbase = blockIdx.x * 256 + wv * 32;
  const int colbase = blockIdx.y * 64;
  const int l15     = lane & 15;
  const int khalf   = (lane >> 4) << 1;

  const float* __restrict__ tr0 = T1 + (size_t)(rowbase + l15) * RANK;
  const float* __restrict__ tr1 = tr0 + (size_t)16 * RANK;
  const float* __restrict__ c0p = C + (size_t)(colbase + l15 +  0) * RANK;
  const float* __restrict__ c1p = C + (size_t)(colbase + l15 + 16) * RANK;
  const float* __restrict__ c2p = C + (size_t)(colbase + l15 + 32) * RANK;
  const float* __restrict__ c3p = C + (size_t)(colbase + l15 + 48) * RANK;

  v8f a00 = {}, a01 = {}, a02 = {}, a03 = {};
  v8f a10 = {}, a11 = {}, a12 = {}, a13 = {};

#pragma unroll
  for (int k = 0; k < RANK; k += 4) {
    const int kk = k + khalf;
    v2f af0 = *(const v2f*)(tr0 + kk);
    v2f af1 = *(const v2f*)(tr1 + kk);
    v2f bf0 = *(const v2f*)(c0p + kk);
    v2f bf1 = *(const v2f*)(c1p + kk);
    v2f bf2 = *(const v2f*)(c2p + kk);
    v2f bf3 = *(const v2f*)(c3p + kk);
    a00 = __builtin_amdgcn_wmma_f32_16x16x4_f32(false, af0, false, bf0, (short)0, a00, false, false);
    a01 = __builtin_amdgcn_wmma_f32_16x16x4_f32(false, af0, false, bf1, (short)0, a01, false, false);
    a02 = __builtin_amdgcn_wmma_f32_16x16x4_f32(false, af0, false, bf2, (short)0, a02, false, false);
    a03 = __builtin_amdgcn_wmma_f32_16x16x4_f32(false, af0, false, bf3, (short)0, a03, false, false);
    a10 = __builtin_amdgcn_wmma_f32_16x16x4_f32(false, af1, false, bf0, (short)0, a10, false, false);
    a11 = __builtin_amdgcn_wmma_f32_16x16x4_f32(false, af1, false, bf1, (short)0, a11, false, false);
    a12 = __builtin_amdgcn_wmma_f32_16x16x4_f32(false, af1, false, bf2, (short)0, a12, false, false);
    a13 = __builtin_amdgcn_wmma_f32_16x16x4_f32(false, af1, false, bf3, (short)0, a13, false, false);
  }

  float* __restrict__ o0 = out + (size_t)(rowbase + (lane >> 4) * 8) * D_LARGE + colbase + l15;
  float* __restrict__ o1 = o0 + (size_t)16 * D_LARGE;
#pragma unroll
  for (int v = 0; v < 8; ++v) {
    o0[(size_t)v * D_LARGE +  0] = a00[v];
    o0[(size_t)v * D_LARGE + 16] = a01[v];
    o0[(size_t)v * D_LARGE + 32] = a02[v];
    o0[(size_t)v * D_LARGE + 48] = a03[v];
    o1[(size_t)v * D_LARGE +  0] = a10[v];
    o1[(size_t)v * D_LARGE + 16] = a11[v];
    o1[(size_t)v * D_LARGE + 32] = a12[v];
    o1[(size_t)v * D_LARGE + 48] = a13[v];
  }
}

// ---------------------------------------------------------------------------

extern "C" void kernel_launch(void* const* d_in, const int* in_sizes, int n_in,
                              void* d_out, int out_size, void* d_ws, size_t ws_size,
                              hipStream_t stream) {
  const float* x = (const float*)d_in[0];   // [4,2048,4096]
  const float* A = (const float*)d_in[1];   // [4096,64]
  const float* B = (const float*)d_in[2];   // [64,768]
  const float* W = (const float*)d_in[3];   // [768,768]
  float* out = (float*)d_out;               // [4,2048,4096]

  // workspace carve-out (floats): ~21.3 MB total
  float* V  = (float*)d_ws;                      // 64*768
  float* M  = V  + (size_t)RANK * D_SMALL;       // 64*64
  float* C  = M  + (size_t)RANK * RANK;          // 4096*64
  float* At = C  + (size_t)D_LARGE * RANK;       // 64*4096
  float* T1 = At + (size_t)RANK * D_LARGE;       // 8192*64
  float* P  = T1 + (size_t)NROWS * RANK;         // 8*8192*64

  k_bw <<<(RANK * D_SMALL) / 256, 256, 0, stream>>>(B, W, V);
  k_m  <<<(RANK * RANK) / 256, 256, 0, stream>>>(B, V, M);
  k_c  <<<(D_LARGE * RANK) / 256, 256, 0, stream>>>(A, M, C, At);
  k_xa <<<dim3(NROWS / 256, KSPLIT), 256, 0, stream>>>(x, At, P);
  k_red<<<(NROWS * RANK) / 256, 256, 0, stream>>>(P, T1);
  k_out<<<dim3(NROWS / 256, D_LARGE / 64), 256, 0, stream>>>(T1, C, out);
}